// Model_60713657697058
// MI455X (gfx1250) — compile-verified
//
#include <hip/hip_runtime.h>

// spence(x) = Li2(1-x), Cephes rational approximation, branchless.
// HBM-bound: 537 MB @ 23.3 TB/s ~ 23 us floor. Single-use stream > 192 MB L2
// -> non-temporal B128 loads/stores. Polynomials + corrections in packed f32
// (V_PK_FMA_F32 etc.), transcendentals/selects per-lane.
//
// NaN/special-point dataflow (matches the jnp.where chain in the reference):
//  - x <  0 : lo path selected, v_log_f32(x<0) = NaN -> corr_lo = NaN  (no
//             explicit NaN select needed)
//  - x == 0 : lx guard (xt==0 -> log(1)=0), l1 = log(1)=0, y=+0 -> PI2_6 exact
//  - x == 1 : w=0 -> y=-0 (== 0.0), corrections not selected
//  - x >  2 : BOTH corrections fire (xt=1/x<0.5 => lo true), and the second
//             must consume the lo-corrected y (Cephes order).

typedef float v4f __attribute__((ext_vector_type(4)));
typedef float v2f __attribute__((ext_vector_type(2)));

__device__ __forceinline__ v2f v2s(float c) { v2f v = {c, c}; return v; }

// ---- packed 2-lane spence ------------------------------------------------
__device__ __forceinline__ v2f spence2(v2f x) {
    const float PI2_6 = 1.64493406684822643647f;   // pi^2/6
    const float LN2   = 0.69314718055994530942f;

    // one v_rcp_f32 per lane serves both the big (x>2 -> xt=1/x) and the
    // hi (xt>1.5, only reachable with xt==x -> w=1/x-1) reduction branches
    v2f r;
    r.x = __builtin_amdgcn_rcpf(x.x);
    r.y = __builtin_amdgcn_rcpf(x.y);

    bool big0 = x.x > 2.0f, big1 = x.y > 2.0f;
    v2f xt;
    xt.x = big0 ? r.x : x.x;
    xt.y = big1 ? r.y : x.y;
    bool hi0 = xt.x > 1.5f, hi1 = xt.y > 1.5f;
    bool lo0 = xt.x < 0.5f, lo1 = xt.y < 0.5f;
    v2f w;
    w.x = hi0 ? (r.x - 1.0f) : (lo0 ? -xt.x : xt.x - 1.0f);
    w.y = hi1 ? (r.y - 1.0f) : (lo1 ? -xt.y : xt.y - 1.0f);

    // polevl(w, A) — packed Horner (v_pk_fma_f32)
    v2f pa = v2s(4.65128586073990045278e-5f);
    pa = __builtin_elementwise_fma(pa, w, v2s(7.31589045238094711071e-3f));
    pa = __builtin_elementwise_fma(pa, w, v2s(1.33847639578309018650e-1f));
    pa = __builtin_elementwise_fma(pa, w, v2s(8.79691311754530315341e-1f));
    pa = __builtin_elementwise_fma(pa, w, v2s(2.71149851196553469920e+0f));
    pa = __builtin_elementwise_fma(pa, w, v2s(4.25697156008121755724e+0f));
    pa = __builtin_elementwise_fma(pa, w, v2s(3.29771340985225106936e+0f));
    pa = __builtin_elementwise_fma(pa, w, v2s(1.00000000000000000126e+0f));

    // polevl(w, B) — B(w) in [1, ~5.03], so 1-ulp v_rcp_f32 is safe
    v2f pb = v2s(6.90990488912553276999e-4f);
    pb = __builtin_elementwise_fma(pb, w, v2s(2.54043763932544379113e-2f));
    pb = __builtin_elementwise_fma(pb, w, v2s(2.82974860602568089943e-1f));
    pb = __builtin_elementwise_fma(pb, w, v2s(1.41172597751831069617e+0f));
    pb = __builtin_elementwise_fma(pb, w, v2s(3.63800533345137075418e+0f));
    pb = __builtin_elementwise_fma(pb, w, v2s(5.03278880143316990390e+0f));
    pb = __builtin_elementwise_fma(pb, w, v2s(4.92478529773146006102e+0f));
    pb = __builtin_elementwise_fma(pb, w, v2s(1.00000000000000000000e+0f));

    v2f ipb;
    ipb.x = __builtin_amdgcn_rcpf(pb.x);
    ipb.y = __builtin_amdgcn_rcpf(pb.y);
    v2f y = (-w) * pa * ipb;                       // packed muls

    // logs: lx guarded only against xt==0 (x<0 must yield NaN via log(neg));
    // l1 unguarded — only consumed when lo is selected (xt<0.5 => 1-xt>0.5)
    v2f omx = v2s(1.0f) - xt;                      // packed sub
    v2f lx, l1;
    lx.x = __builtin_amdgcn_logf(xt.x == 0.0f ? 1.0f : xt.x);
    lx.y = __builtin_amdgcn_logf(xt.y == 0.0f ? 1.0f : xt.y);
    l1.x = __builtin_amdgcn_logf(omx.x);
    l1.y = __builtin_amdgcn_logf(omx.y);
    lx = lx * v2s(LN2);
    l1 = l1 * v2s(LN2);

    // flag1 correction (lo) FIRST — Cephes order
    v2f corr_lo = __builtin_elementwise_fma(-lx, l1, v2s(PI2_6)) - y;
    y.x = lo0 ? corr_lo.x : y.x;
    y.y = lo1 ? corr_lo.y : y.y;

    // flag2 correction (big|hi) consumes the UPDATED y (x>2 hits both flags)
    v2f corr_hi = __builtin_elementwise_fma(lx * v2s(-0.5f), lx, -y);
    y.x = (big0 || hi0) ? corr_hi.x : y.x;
    y.y = (big1 || hi1) ? corr_hi.y : y.y;
    return y;
}

// ---- scalar path for the tail -------------------------------------------
__device__ __forceinline__ float spence1(float x) {
    v2f v = {x, x};
    v2f r = spence2(v);
    return r.x;
}

extern "C" __global__ __launch_bounds__(256)
void spence_kernel(const float* __restrict__ in, float* __restrict__ out,
                   long long n) {
    long long t    = (long long)blockIdx.x * 256 + threadIdx.x;
    long long base = t << 3;                       // 8 elements per thread
    if (base + 7 < n) {
        v4f a = __builtin_nontemporal_load((const v4f*)in + 2 * t);
        v4f b = __builtin_nontemporal_load((const v4f*)in + 2 * t + 1);
        v2f p0 = {a.x, a.y}; v2f p1 = {a.z, a.w};
        v2f p2 = {b.x, b.y}; v2f p3 = {b.z, b.w};
        v2f r0 = spence2(p0);
        v2f r1 = spence2(p1);
        v2f r2 = spence2(p2);
        v2f r3 = spence2(p3);
        v4f oa = {r0.x, r0.y, r1.x, r1.y};
        v4f ob = {r2.x, r2.y, r3.x, r3.y};
        __builtin_nontemporal_store(oa, (v4f*)out + 2 * t);
        __builtin_nontemporal_store(ob, (v4f*)out + 2 * t + 1);
    } else if (base < n) {
        for (long long j = base; j < n; ++j)
            out[j] = spence1(in[j]);
    }
}

extern "C" void kernel_launch(void* const* d_in, const int* in_sizes, int n_in,
                              void* d_out, int out_size, void* d_ws, size_t ws_size,
                              hipStream_t stream) {
    const float* in  = (const float*)d_in[0];
    float*       out = (float*)d_out;
    long long n = (long long)in_sizes[0];          // 64*1024*1024
    long long nthreads = (n + 7) >> 3;
    long long blocks   = (nthreads + 255) / 256;
    if (blocks < 1) blocks = 1;
    spence_kernel<<<(dim3)(unsigned)blocks, 256, 0, stream>>>(in, out, n);
}